// YOLOF_28140625723713
// MI455X (gfx1250) — compile-verified
//
#include <hip/hip_runtime.h>
#include <hip/hip_bf16.h>
#include <math.h>

// ---------------- problem constants ----------------
constexpr int B_    = 32;
constexpr int A_    = 24576;          // 64*64*6 anchors
constexpr int C_    = 80;
constexpr int AC_   = A_ * C_;        // 1966080 scores per image
constexpr int K_    = 1000;
constexpr int MAXD_ = 100;
constexpr float SCALE_CLAMP_F = 4.135166556742356f;   // log(1000/16)
constexpr float NMS_TH_  = 0.6f;
constexpr float SCORE_TH_ = 0.05f;

// ---------------- tiling ----------------
constexpr int CHUNKS = 8;             // blocks per image for streaming passes
constexpr int SEG    = AC_ / CHUNKS;  // 245760
constexpr int TILE   = 4096;          // elements per TDM tile (16KB)
constexpr int NTILES = SEG / TILE;    // 60
constexpr unsigned TIECAP = 8192;

// ---------------- workspace layout (u32 units) ----------------
constexpr unsigned OFF_HIST    = 0;                                   // 32*2048
constexpr unsigned OFF_CNT_HI  = OFF_HIST   + 32u * 2048u;            // 32
constexpr unsigned OFF_CNT_TIE = OFF_CNT_HI + 32u;                    // 32
constexpr unsigned OFF_META    = OFF_CNT_TIE + 32u;                   // 32*8 {t1,hiCount,need}
constexpr unsigned OFF_HIBUF   = OFF_META   + 32u * 8u;               // 32*1024*2
constexpr unsigned OFF_TIEA    = OFF_HIBUF  + 32u * 1024u * 2u;
constexpr unsigned OFF_TIEB    = OFF_TIEA   + 32u * TIECAP * 2u;      // end ~4.7MB

__device__ __forceinline__ unsigned umin_(unsigned a, unsigned b) { return a < b ? a : b; }

// monotonic float->u32 key (larger float => larger key)
__device__ __forceinline__ unsigned f2key(float x) {
  unsigned u = __float_as_uint(x);
  return (u & 0x80000000u) ? ~u : (u | 0x80000000u);
}
__device__ __forceinline__ float key2f(unsigned k) {
  unsigned u = (k & 0x80000000u) ? (k & 0x7FFFFFFFu) : ~k;
  return __uint_as_float(u);
}

// ---------------- CDNA5 Tensor Data Mover staging ----------------
#ifdef __AMDGCN__
#  if __has_builtin(__builtin_amdgcn_tensor_load_to_lds) && __has_builtin(__builtin_amdgcn_s_wait_tensorcnt)
#    define USE_TDM 1
#    warning "CDNA5: TDM path ENABLED (tensor_load_to_lds + s_wait_tensorcnt)"
#  else
#    warning "CDNA5: TDM builtins NOT available - falling back to plain global loads"
#  endif
#endif

#ifdef USE_TDM
typedef unsigned int u32x4 __attribute__((ext_vector_type(4)));
typedef int          i32x4 __attribute__((ext_vector_type(4)));
typedef int          i32x8 __attribute__((ext_vector_type(8)));

// Low 32 bits of a flat pointer to LDS are the workgroup-relative LDS byte offset.
__device__ __forceinline__ unsigned lds_byte_off(const void* p) {
  return (unsigned)(unsigned long long)p;
}

// 1D contiguous tile of `nelem` f32 from global -> LDS via TDM (wave-uniform call).
__device__ __forceinline__ void tdm_load_1d(const float* gsrc, unsigned ldsoff, int nelem) {
  unsigned long long ga = (unsigned long long)(const void*)gsrc;
  u32x4 g0;
  g0.x = 1u;                                                 // count=1, user descriptor
  g0.y = ldsoff;                                             // lds_addr [63:32]
  g0.z = (unsigned)(ga & 0xFFFFFFFFull);                     // global_addr lo
  g0.w = (unsigned)((ga >> 32) & 0x1FFFFFFull) | (2u << 30); // addr hi + type=2
  i32x8 g1;
  unsigned n = (unsigned)nelem;
  g1[0] = (int)(2u << 16);                 // data_size=4B, no mask/flags
  g1[1] = (int)((n & 0xFFFFu) << 16);      // tensor_dim0 lo16 (bits 79:48)
  g1[2] = (int)((n >> 16) & 0xFFFFu);      // tensor_dim0 hi16; tensor_dim1=0
  g1[3] = (int)((n & 0xFFFFu) << 16);      // tile_dim0 (bits 127:112)
  g1[4] = 0; g1[5] = 0; g1[6] = 0; g1[7] = 0; // tile_dim1/2=0, strides=0
  i32x4 z4 = {0, 0, 0, 0};
#if defined(__clang_major__) && (__clang_major__ >= 23)
#  warning "CDNA5: using 6-arg tensor_load_to_lds (clang>=23)"
  i32x8 z8 = {0, 0, 0, 0, 0, 0, 0, 0};
  __builtin_amdgcn_tensor_load_to_lds(g0, g1, z4, z4, z8, 0);
#else
#  warning "CDNA5: using 5-arg tensor_load_to_lds (clang<23)"
  __builtin_amdgcn_tensor_load_to_lds(g0, g1, z4, z4, 0);
#endif
}
#endif // USE_TDM

// ---------------- kernel: 2048-bin key histogram (TDM-staged stream) ----------------
// First in file so the disasm snippet shows the TDM pipeline.
__global__ __launch_bounds__(256) void yolof_hist(const float* __restrict__ logits,
                                                  unsigned* __restrict__ ws) {
  __shared__ unsigned lhist[2048];
#ifdef USE_TDM
  __shared__ float stage[2][TILE];
#endif
  const int img   = blockIdx.x / CHUNKS;
  const int chunk = blockIdx.x % CHUNKS;
  const float* src = logits + (size_t)img * AC_ + (size_t)chunk * SEG;

  for (int i = threadIdx.x; i < 2048; i += 256) lhist[i] = 0u;
  __syncthreads();

#ifdef USE_TDM
  const bool issuer = (threadIdx.x >> 5) == 0;  // wave 0 issues DMAs (wave-uniform)
  if (issuer) tdm_load_1d(src, lds_byte_off(&stage[0][0]), TILE);
  for (int t = 0; t < NTILES; ++t) {
    const int cur = t & 1;
    if (issuer) {
      if (t + 1 < NTILES) {
        tdm_load_1d(src + (size_t)(t + 1) * TILE, lds_byte_off(&stage[cur ^ 1][0]), TILE);
        __builtin_amdgcn_s_wait_tensorcnt(1);   // oldest (tile t) complete
      } else {
        __builtin_amdgcn_s_wait_tensorcnt(0);
      }
    }
    __syncthreads();
    for (int i = threadIdx.x; i < TILE; i += 256) {
      unsigned key = f2key(stage[cur][i]);
      atomicAdd(&lhist[key >> 21], 1u);
    }
    __syncthreads();   // buffer refilled next iteration
  }
#else
  for (int i = threadIdx.x; i < SEG; i += 256) {
    unsigned key = f2key(src[i]);
    atomicAdd(&lhist[key >> 21], 1u);
  }
#endif
  __syncthreads();
  unsigned* gh = ws + OFF_HIST + (unsigned)img * 2048u;
  for (int i = threadIdx.x; i < 2048; i += 256)
    if (lhist[i]) atomicAdd(&gh[i], lhist[i]);
}

// ---------------- kernel: zero histograms + counters ----------------
__global__ void yolof_init(unsigned* __restrict__ ws) {
  unsigned n = OFF_META;  // hist + both counter arrays
  for (unsigned i = blockIdx.x * blockDim.x + threadIdx.x; i < n; i += gridDim.x * blockDim.x)
    ws[i] = 0u;
}

// ---------------- kernel: find level-1 threshold bin per image ----------------
__global__ void yolof_thresh(unsigned* __restrict__ ws) {
  if (threadIdx.x != 0) return;
  const int img = blockIdx.x;
  const unsigned* gh = ws + OFF_HIST + (unsigned)img * 2048u;
  unsigned cum = 0, t1 = 0, hi = 0;
  for (int b = 2047; b >= 0; --b) {
    unsigned c = gh[b];
    if (cum + c >= (unsigned)K_) { t1 = (unsigned)b; hi = cum; break; }
    cum += c;
  }
  unsigned* meta = ws + OFF_META + (unsigned)img * 8u;
  meta[0] = t1; meta[1] = hi; meta[2] = (unsigned)K_ - hi;
}

// ---------------- kernel: collect definite + tie candidates (TDM-staged) ----------------
__global__ __launch_bounds__(256) void yolof_collect(const float* __restrict__ logits,
                                                     unsigned* __restrict__ ws) {
  const int img   = blockIdx.x / CHUNKS;
  const int chunk = blockIdx.x % CHUNKS;
  const float* src = logits + (size_t)img * AC_ + (size_t)chunk * SEG;
  const unsigned t1 = ws[OFF_META + (unsigned)img * 8u];
  unsigned* hiCnt  = ws + OFF_CNT_HI + img;
  unsigned* tieCnt = ws + OFF_CNT_TIE + img;
  unsigned* hiBuf  = ws + OFF_HIBUF + (unsigned)img * 1024u * 2u;
  unsigned* tieA   = ws + OFF_TIEA + (unsigned)img * TIECAP * 2u;
  const unsigned base_idx = (unsigned)chunk * SEG;

#ifdef USE_TDM
  __shared__ float stage[2][TILE];
  const bool issuer = (threadIdx.x >> 5) == 0;
  if (issuer) tdm_load_1d(src, lds_byte_off(&stage[0][0]), TILE);
  for (int t = 0; t < NTILES; ++t) {
    const int cur = t & 1;
    if (issuer) {
      if (t + 1 < NTILES) {
        tdm_load_1d(src + (size_t)(t + 1) * TILE, lds_byte_off(&stage[cur ^ 1][0]), TILE);
        __builtin_amdgcn_s_wait_tensorcnt(1);
      } else {
        __builtin_amdgcn_s_wait_tensorcnt(0);
      }
    }
    __syncthreads();
    const unsigned tbase = base_idx + (unsigned)t * TILE;
    for (int i = threadIdx.x; i < TILE; i += 256) {
      unsigned key = f2key(stage[cur][i]);
      unsigned b = key >> 21;
      if (b > t1) {
        unsigned p = atomicAdd(hiCnt, 1u);
        if (p < 1024u) { hiBuf[p * 2] = key; hiBuf[p * 2 + 1] = tbase + (unsigned)i; }
      } else if (b == t1) {
        unsigned p = atomicAdd(tieCnt, 1u);
        if (p < TIECAP) { tieA[p * 2] = key; tieA[p * 2 + 1] = tbase + (unsigned)i; }
      }
    }
    __syncthreads();
  }
#else
  for (int i = threadIdx.x; i < SEG; i += 256) {
    unsigned key = f2key(src[i]);
    unsigned b = key >> 21;
    if (b > t1) {
      unsigned p = atomicAdd(hiCnt, 1u);
      if (p < 1024u) { hiBuf[p * 2] = key; hiBuf[p * 2 + 1] = base_idx + (unsigned)i; }
    } else if (b == t1) {
      unsigned p = atomicAdd(tieCnt, 1u);
      if (p < TIECAP) { tieA[p * 2] = key; tieA[p * 2 + 1] = base_idx + (unsigned)i; }
    }
  }
#endif
}

// ---------------- kernel: exact select + sort + decode + NMS + emit ----------------
__global__ __launch_bounds__(1024) void yolof_finish(const float* __restrict__ deltas,
                                                     const float* __restrict__ anchors,
                                                     unsigned* __restrict__ ws,
                                                     float* __restrict__ out) {
  const int img = blockIdx.x;
  const unsigned tid = threadIdx.x;

  __shared__ unsigned long long comp[1024];
  __shared__ unsigned hist[2048];
  __shared__ float bx[1024][4];
  __shared__ float sc[1024];
  __shared__ unsigned short clsA[1024];
  __shared__ unsigned eq[1024];
  __shared__ unsigned supp[32];
  __shared__ unsigned s_selCnt, s_nextCnt, s_t, s_hi, s_need, s_m;

  unsigned* meta   = ws + OFF_META + (unsigned)img * 8u;
  const unsigned t1      = meta[0];
  const unsigned hiCount = meta[1];
  unsigned need          = meta[2];
  const unsigned tieN = umin_(ws[OFF_CNT_TIE + img], TIECAP);
  unsigned* hiBuf = ws + OFF_HIBUF + (unsigned)img * 1024u * 2u;
  unsigned* bufA  = ws + OFF_TIEA + (unsigned)img * TIECAP * 2u;
  unsigned* bufB  = ws + OFF_TIEB + (unsigned)img * TIECAP * 2u;

  // --- seed selection with definite entries ---
  comp[tid] = 0ull;
  if (tid == 0) s_selCnt = hiCount;
  __syncthreads();
  for (unsigned i = tid; i < hiCount && i < 1024u; i += 1024u) {
    unsigned key = hiBuf[i * 2], idx = hiBuf[i * 2 + 1];
    comp[i] = ((unsigned long long)key << 32) | (0xFFFFFFFFu - idx);
  }
  __syncthreads();

  // --- refine level 2: bits (key>>10)&0x7FF over tie list ---
  for (unsigned i = tid; i < 2048u; i += 1024u) hist[i] = 0u;
  __syncthreads();
  for (unsigned i = tid; i < tieN; i += 1024u)
    atomicAdd(&hist[(bufA[i * 2] >> 10) & 0x7FFu], 1u);
  __syncthreads();
  if (tid == 0) {
    unsigned cum = 0, t = 0, hi = 0;
    for (int b = 2047; b >= 0; --b) {
      unsigned c = hist[b];
      if (cum + c >= need) { t = (unsigned)b; hi = cum; break; }
      cum += c;
    }
    s_t = t; s_hi = hi; s_nextCnt = 0u;
  }
  __syncthreads();
  const unsigned t2 = s_t;
  for (unsigned i = tid; i < tieN; i += 1024u) {
    unsigned key = bufA[i * 2], idx = bufA[i * 2 + 1];
    unsigned b = (key >> 10) & 0x7FFu;
    if (b > t2) {
      unsigned p = atomicAdd(&s_selCnt, 1u);
      if (p < 1024u) comp[p] = ((unsigned long long)key << 32) | (0xFFFFFFFFu - idx);
    } else if (b == t2) {
      unsigned p = atomicAdd(&s_nextCnt, 1u);
      if (p < TIECAP) { bufB[p * 2] = key; bufB[p * 2 + 1] = idx; }
    }
  }
  __syncthreads();
  if (tid == 0) { s_need = need - s_hi; s_m = umin_(s_nextCnt, TIECAP); }
  __syncthreads();
  need = s_need;
  const unsigned n2 = s_m;

  // --- refine level 3: bits key&0x3FF ---
  for (unsigned i = tid; i < 2048u; i += 1024u) hist[i] = 0u;
  __syncthreads();
  for (unsigned i = tid; i < n2; i += 1024u)
    atomicAdd(&hist[bufB[i * 2] & 0x3FFu], 1u);
  __syncthreads();
  if (tid == 0) {
    unsigned cum = 0, t = 0, hi = 0;
    for (int b = 1023; b >= 0; --b) {
      unsigned c = hist[b];
      if (cum + c >= need) { t = (unsigned)b; hi = cum; break; }
      cum += c;
    }
    s_t = t; s_hi = hi; s_nextCnt = 0u;
  }
  __syncthreads();
  const unsigned t3 = s_t;
  for (unsigned i = tid; i < n2; i += 1024u) {
    unsigned key = bufB[i * 2], idx = bufB[i * 2 + 1];
    unsigned b = key & 0x3FFu;
    if (b > t3) {
      unsigned p = atomicAdd(&s_selCnt, 1u);
      if (p < 1024u) comp[p] = ((unsigned long long)key << 32) | (0xFFFFFFFFu - idx);
    } else if (b == t3) {
      unsigned p = atomicAdd(&s_nextCnt, 1u);
      if (p < 1024u) eq[p] = idx;
    }
  }
  __syncthreads();
  if (tid == 0) { s_need = need - s_hi; s_m = umin_(s_nextCnt, 1024u); }
  __syncthreads();
  need = s_need;
  const unsigned m = s_m;

  // --- exact-key ties: smallest indices win (odd-even sort ascending) ---
  for (unsigned p = 0; p < m; ++p) {
    unsigned i = 2u * tid + (p & 1u);
    if (i + 1u < m) {
      unsigned a = eq[i], b = eq[i + 1];
      if (a > b) { eq[i] = b; eq[i + 1] = a; }
    }
    __syncthreads();
  }
  {
    const unsigned Tkey = (t1 << 21) | (t2 << 10) | t3;
    const unsigned base = s_selCnt;
    if (tid < need && tid < m) {
      unsigned p = base + tid;
      if (p < 1024u) comp[p] = ((unsigned long long)Tkey << 32) | (0xFFFFFFFFu - eq[tid]);
    }
  }
  __syncthreads();

  // --- bitonic sort 1024 entries, descending by composite ---
  for (unsigned k = 2; k <= 1024u; k <<= 1)
    for (unsigned j = k >> 1; j > 0; j >>= 1) {
      __syncthreads();
      unsigned i = tid, ixj = i ^ j;
      if (ixj > i) {
        unsigned long long a = comp[i], b = comp[ixj];
        bool up = ((i & k) == 0u);
        if ((a < b) == up) { comp[i] = b; comp[ixj] = a; }
      }
    }
  __syncthreads();

  // --- decode score, class, box ---
  {
    unsigned long long c = comp[tid];
    unsigned key = (unsigned)(c >> 32);
    unsigned idx = 0xFFFFFFFFu - (unsigned)(c & 0xFFFFFFFFull);
    float logit = key2f(key);
    sc[tid] = 1.0f / (1.0f + expf(-logit));
    unsigned aidx = idx / (unsigned)C_;
    clsA[tid] = (unsigned short)(idx % (unsigned)C_);
    if (tid < (unsigned)K_ && aidx < (unsigned)A_) {
      const float* an = anchors + (size_t)aidx * 4;
      const float* de = deltas + ((size_t)img * A_ + aidx) * 4;
      float x1 = an[0], y1 = an[1], x2 = an[2], y2 = an[3];
      float w = x2 - x1, h = y2 - y1;
      float cx = x1 + 0.5f * w, cy = y1 + 0.5f * h;
      float dx = de[0], dy = de[1];
      float dw = fminf(de[2], SCALE_CLAMP_F), dh = fminf(de[3], SCALE_CLAMP_F);
      float pcx = dx * w + cx, pcy = dy * h + cy;
      float pw = expf(dw) * w, ph = expf(dh) * h;
      bx[tid][0] = pcx - 0.5f * pw; bx[tid][1] = pcy - 0.5f * ph;
      bx[tid][2] = pcx + 0.5f * pw; bx[tid][3] = pcy + 0.5f * ph;
    } else {
      bx[tid][0] = bx[tid][1] = bx[tid][2] = bx[tid][3] = 0.0f;
    }
  }
  if (tid < 32u) supp[tid] = 0u;
  __syncthreads();
  {
    bool valid = (tid < (unsigned)K_) && (sc[tid] > SCORE_TH_);
    if (!valid) atomicOr(&supp[tid >> 5], 1u << (tid & 31u));
  }

  // --- greedy NMS on class-offset boxes (faithful to reference) ---
  for (int i = 0; i < K_; ++i) {
    __syncthreads();
    bool alive = ((supp[i >> 5] >> (i & 31)) & 1u) == 0u;   // uniform read
    if (alive) {
      int j = (int)tid;
      if (j > i && j < K_) {
        float oi = (float)clsA[i] * 10000.0f;
        float oj = (float)clsA[j] * 10000.0f;
        float xx1 = fmaxf(bx[i][0] + oi, bx[j][0] + oj);
        float yy1 = fmaxf(bx[i][1] + oi, bx[j][1] + oj);
        float xx2 = fminf(bx[i][2] + oi, bx[j][2] + oj);
        float yy2 = fminf(bx[i][3] + oi, bx[j][3] + oj);
        float iw = fmaxf(xx2 - xx1, 0.0f), ih = fmaxf(yy2 - yy1, 0.0f);
        float inter = iw * ih;
        float a1 = (bx[i][2] - bx[i][0]) * (bx[i][3] - bx[i][1]);
        float a2 = (bx[j][2] - bx[j][0]) * (bx[j][3] - bx[j][1]);
        float iou = inter / (a1 + a2 - inter + 1e-9f);
        if (iou > NMS_TH_) atomicOr(&supp[j >> 5], 1u << (j & 31u));
      }
    }
  }
  __syncthreads();

  // --- prefix-scan kept flags, emit top-100 (kept first, earliest non-kept fill) ---
  unsigned keptFlag = (tid < (unsigned)K_) && (((supp[tid >> 5] >> (tid & 31u)) & 1u) == 0u);
  hist[tid] = keptFlag;
  __syncthreads();
  for (unsigned off = 1; off < 1024u; off <<= 1) {
    unsigned v = hist[tid];
    unsigned add = (tid >= off) ? hist[tid - off] : 0u;
    __syncthreads();
    hist[tid] = v + add;
    __syncthreads();
  }
  const unsigned incl = hist[tid];
  const unsigned excl = incl - keptFlag;
  const unsigned totKept = hist[1023];

  float* det = out + (size_t)img * MAXD_ * 5;
  float* clsOut = out + (size_t)B_ * MAXD_ * 5 + (size_t)img * MAXD_;
  if (tid < (unsigned)K_) {
    if (keptFlag) {
      if (excl < (unsigned)MAXD_) {
        unsigned s = excl;
        det[s * 5 + 0] = bx[tid][0]; det[s * 5 + 1] = bx[tid][1];
        det[s * 5 + 2] = bx[tid][2]; det[s * 5 + 3] = bx[tid][3];
        det[s * 5 + 4] = sc[tid];
        clsOut[s] = (float)clsA[tid];
      }
    } else {
      unsigned s = totKept + (tid - excl);
      if (s < (unsigned)MAXD_) {
        det[s * 5 + 0] = bx[tid][0]; det[s * 5 + 1] = bx[tid][1];
        det[s * 5 + 2] = bx[tid][2]; det[s * 5 + 3] = bx[tid][3];
        det[s * 5 + 4] = 0.0f;
        clsOut[s] = (float)clsA[tid];
      }
    }
  }
}

// ---------------- launcher ----------------
extern "C" void kernel_launch(void* const* d_in, const int* in_sizes, int n_in,
                              void* d_out, int out_size, void* d_ws, size_t ws_size,
                              hipStream_t stream) {
  (void)in_sizes; (void)n_in; (void)out_size; (void)ws_size;
  const float* logits  = (const float*)d_in[0];   // [B, A, C]
  const float* deltas  = (const float*)d_in[1];   // [B, A, 4]
  const float* anchors = (const float*)d_in[2];   // [A, 4]
  float* out = (float*)d_out;                     // det [B,100,5] then cls [B,100]
  unsigned* ws = (unsigned*)d_ws;

  yolof_init<<<256, 256, 0, stream>>>(ws);
  yolof_hist<<<B_ * CHUNKS, 256, 0, stream>>>(logits, ws);
  yolof_thresh<<<B_, 32, 0, stream>>>(ws);
  yolof_collect<<<B_ * CHUNKS, 256, 0, stream>>>(logits, ws);
  yolof_finish<<<B_, 1024, 0, stream>>>(deltas, anchors, ws, out);
}